// RandomMask_72567767433410
// MI455X (gfx1250) — compile-verified
//
#include <hip/hip_runtime.h>

// out[b,c,h,w] = x[b,c,h,w] * (1 - mask[b,0,h,w])   (MASK_VALUE == 0.0)
// x: (64,3,512,512) f32, mask: (64,1,512,512) f32, out: same as x.
//
// Pure HBM-bandwidth kernel: ~470 MB traffic, ~0.1 FLOP/byte -> memory bound.
// Strategy: b128 (float4) loads/stores, non-temporal hints (streaming data,
// don't pollute 192MB L2), and in-register mask reuse across the 3 channels.

typedef float v4f __attribute__((ext_vector_type(4)));

#define HW4_LOG2 16u           // (512*512)/4 = 65536 = 2^16 float4s per (b, channel) plane
#define HW4      (1u << HW4_LOG2)
#define NCH      3u

__global__ __launch_bounds__(256) void RandomMask_blend_kernel(
    const float* __restrict__ x,
    const float* __restrict__ mask,
    float* __restrict__ out)
{
    // One thread per mask-float4: tid in [0, B*HW4) = [0, 64*65536)
    const unsigned tid = blockIdx.x * 256u + threadIdx.x;
    const unsigned b   = tid >> HW4_LOG2;      // batch index
    const unsigned hw4 = tid & (HW4 - 1u);     // float4 index within the HxW plane

    const v4f* __restrict__ mp = (const v4f*)mask;
    const v4f* __restrict__ xp = (const v4f*)x;
    v4f*       __restrict__ op = (v4f*)out;

    // Mask plane is contiguous over (B,H,W): float4 index == tid.
    const v4f m = __builtin_nontemporal_load(mp + tid);
    v4f w;
    w.x = 1.0f - m.x;
    w.y = 1.0f - m.y;
    w.z = 1.0f - m.z;
    w.w = 1.0f - m.w;

    // x/out float4 index for channel c: b*3*HW4 + c*HW4 + hw4
    const unsigned base = b * (NCH * HW4) + hw4;

    const v4f x0 = __builtin_nontemporal_load(xp + base);
    const v4f x1 = __builtin_nontemporal_load(xp + base + HW4);
    const v4f x2 = __builtin_nontemporal_load(xp + base + 2u * HW4);

    __builtin_nontemporal_store(x0 * w, op + base);
    __builtin_nontemporal_store(x1 * w, op + base + HW4);
    __builtin_nontemporal_store(x2 * w, op + base + 2u * HW4);
}

extern "C" void kernel_launch(void* const* d_in, const int* in_sizes, int n_in,
                              void* d_out, int out_size, void* d_ws, size_t ws_size,
                              hipStream_t stream)
{
    (void)in_sizes; (void)n_in; (void)out_size; (void)d_ws; (void)ws_size;

    const float* x    = (const float*)d_in[0];   // (64,3,512,512) f32
    const float* mask = (const float*)d_in[1];   // (64,1,512,512) f32
    float*       out  = (float*)d_out;           // (64,3,512,512) f32

    // 64 batches * 65536 mask-float4s = 4,194,304 threads; 256 threads/block.
    const unsigned total_threads = 64u * HW4;
    const unsigned blocks = total_threads / 256u;   // 16384

    RandomMask_blend_kernel<<<blocks, 256, 0, stream>>>(x, mask, out);
}